// GenesisV2_36730560315755
// MI455X (gfx1250) — compile-verified
//
#include <hip/hip_runtime.h>

// GenesisV2 stick-breaking masks for MI455X (gfx1250, wave32).
// Kernel 1: SemiConv 1x1 via V_WMMA_F32_16X16X4_F32 (fp32 matrix pipe).
// Kernel 2: fused 10-step stick-breaking loop, one workgroup per batch.

typedef float v2f __attribute__((ext_vector_type(2)));
typedef float v8f __attribute__((ext_vector_type(8)));

#define B_   64
#define F_   64
#define C_   8
#define S_   128
#define HW_  (S_ * S_)

// ---------------------------------------------------------------------------
// Kernel 1: colour[b, pix, c] = gate*(sum_f feat[b,f,pix]*W[c,f] + bias[c]) + uv
// One wave per 16-pixel x 16-channel tile (channels 8..15 are padding / masked).
// A (16x4 f32): lane = M%16; lanes 0-15 hold K=0,1 (vgpr .x/.y), 16-31 hold K=2,3.
// B (4x16 f32): lane = N%16; same K split across lane halves.
// D (16x16 f32, v8f): vgpr r, lanes 0-15 -> M=r, lanes 16-31 -> M=r+8; N = lane%16.
// ---------------------------------------------------------------------------
__global__ __launch_bounds__(256) void semiconv_wmma_f32(
    const float* __restrict__ feat,   // [B, F, HW]
    const float* __restrict__ w,      // [C, F]  (conv_w squeezed)
    const float* __restrict__ bias,   // [C]
    const float* __restrict__ gate,   // [1]
    float* __restrict__ colour)       // [B, HW, C]  workspace
{
    const int lane    = threadIdx.x & 31;
    const int waveBlk = threadIdx.x >> 5;
    const int tile    = blockIdx.x * 8 + waveBlk;   // 0 .. B*HW/16 - 1
    const int b       = tile >> 10;                 // HW/16 = 1024 tiles per batch
    const int pixBase = (tile & 1023) << 4;

    const int m  = lane & 15;     // A: row M   | B: col N  | D: col N
    const int hi = lane >> 4;     // selects K pair (A/B) and M half (D)

    const float g = gate[0];
    const float* fb = feat + (size_t)b * F_ * HW_;

    // Branchless B-matrix fetch so EXEC stays all-1s for the WMMAs.
    const int   wrow  = (m < C_) ? m : 0;
    const float wmask = (m < C_) ? 1.f : 0.f;

    v8f acc = {0.f, 0.f, 0.f, 0.f, 0.f, 0.f, 0.f, 0.f};

    #pragma unroll
    for (int kk = 0; kk < F_ / 4; ++kk) {
        const int k0 = kk * 4 + hi * 2;
        v2f a, bf;
        a.x  = fb[(size_t)(k0 + 0) * HW_ + pixBase + m];
        a.y  = fb[(size_t)(k0 + 1) * HW_ + pixBase + m];
        bf.x = w[wrow * F_ + k0 + 0] * wmask;
        bf.y = w[wrow * F_ + k0 + 1] * wmask;
        // (neg_a, A, neg_b, B, c_mod, C, reuse_a, reuse_b)
        acc = __builtin_amdgcn_wmma_f32_16x16x4_f32(
            false, a, false, bf, (short)0, acc, false, false);
    }

    if (m < C_) {
        const float bc = bias[m];
        #pragma unroll
        for (int r = 0; r < 8; ++r) {
            const int pix = pixBase + r + hi * 8;
            const int h   = pix >> 7;          // row (S_=128)
            const int wc  = pix & 127;         // col
            float uv = (m == C_ - 2) ? (-1.f + (float)h  * (2.f / (S_ - 1)))
                     : (m == C_ - 1) ? (-1.f + (float)wc * (2.f / (S_ - 1)))
                     : 0.f;
            colour[((size_t)b * HW_ + pix) * C_ + m] = g * (acc[r] + bc) + uv;
        }
    }
}

// ---------------------------------------------------------------------------
// Kernel 2: fused stick-breaking loop. One 1024-thread workgroup per batch;
// each thread owns 16 pixels (log-scope + rand kept in registers), argmax via
// LDS tree reduction with first-occurrence tie-break, seed broadcast via LDS.
// Colour re-read each step comes from L2 (512 KB / batch).
// ---------------------------------------------------------------------------
__global__ __launch_bounds__(1024) void genesis_steps(
    const float* __restrict__ colour,     // [B, HW, C]
    const float* __restrict__ rand_pixel, // [B, HW]
    const float* __restrict__ log_sigma,  // [1]
    float* __restrict__ out,              // [B, HW, steps+1]
    int steps)
{
    const int b  = blockIdx.x;
    const int t  = threadIdx.x;
    const int K1 = steps + 1;

    const float inv_sigma = expf(-log_sigma[0]);   // 1/sigma = exp(-log_sigma)

    const float* col = colour     + (size_t)b * HW_ * C_;
    const float* rp  = rand_pixel + (size_t)b * HW_;
    float*       o   = out        + (size_t)b * HW_ * K1;

    float ls[16];   // log-scope, per owned pixel
    float rv[16];   // rand_pixel, per owned pixel
    #pragma unroll
    for (int i = 0; i < 16; ++i) {
        rv[i] = rp[t + i * 1024];
        ls[i] = 0.f;
    }

    __shared__ float s_val[1024];
    __shared__ int   s_idx[1024];
    __shared__ float s_seed[C_];

    for (int k = 0; k < steps; ++k) {
        // ---- per-thread argmax of rand * exp(log_scope) (pix ascending ->
        //      strict '>' keeps first occurrence, matching jnp.argmax) ----
        float best = -1.f;
        int   bi   = 0;
        #pragma unroll
        for (int i = 0; i < 16; ++i) {
            const float v = rv[i] * expf(ls[i]);
            if (v > best) { best = v; bi = t + i * 1024; }
        }
        s_val[t] = best;
        s_idx[t] = bi;
        __syncthreads();

        // ---- block tree reduction, lowest index wins ties ----
        for (int s = 512; s > 0; s >>= 1) {
            if (t < s) {
                const float v1 = s_val[t],     v2 = s_val[t + s];
                const int   i1 = s_idx[t],     i2 = s_idx[t + s];
                if (v2 > v1 || (v2 == v1 && i2 < i1)) {
                    s_val[t] = v2;
                    s_idx[t] = i2;
                }
            }
            __syncthreads();
        }

        // ---- broadcast seed colour ----
        if (t < C_) s_seed[t] = col[(size_t)s_idx[0] * C_ + t];
        __syncthreads();
        float sd[C_];
        #pragma unroll
        for (int c = 0; c < C_; ++c) sd[c] = s_seed[c];

        // ---- gaussian mask + scope update ----
        #pragma unroll
        for (int i = 0; i < 16; ++i) {
            const int pix = t + i * 1024;
            const float* cp = col + (size_t)pix * C_;
            float d2 = 0.f;
            #pragma unroll
            for (int c = 0; c < C_; ++c) {
                const float df = cp[c] - sd[c];
                d2 += df * df;
            }
            float alpha = expf(-d2 * inv_sigma);
            alpha = fminf(fmaxf(alpha, 0.01f), 0.99f);
            o[(size_t)pix * K1 + k] = ls[i] + logf(alpha);
            ls[i] += logf(1.f - alpha);
        }
        __syncthreads();
    }

    // remaining scope is the final mask
    #pragma unroll
    for (int i = 0; i < 16; ++i) {
        const int pix = t + i * 1024;
        o[(size_t)pix * K1 + steps] = ls[i];
    }
}

// ---------------------------------------------------------------------------
extern "C" void kernel_launch(void* const* d_in, const int* in_sizes, int n_in,
                              void* d_out, int out_size, void* d_ws, size_t ws_size,
                              hipStream_t stream)
{
    const float* feat  = (const float*)d_in[0];  // [64,64,128,128]
    const float* rp    = (const float*)d_in[1];  // [64,1,128,128]
    const float* convw = (const float*)d_in[2];  // [8,64,1,1]
    const float* convb = (const float*)d_in[3];  // [8]
    const float* gate  = (const float*)d_in[4];  // [1]
    const float* lsig  = (const float*)d_in[5];  // [1]
    // d_in[6] (steps_to_run) is a device int; derive steps from out_size
    // on the host so the launch is deterministic and graph-capture safe.
    const int steps = out_size / (B_ * HW_) - 1;   // == 10

    float* colour = (float*)d_ws;                  // B*HW*C_ floats = 32 MiB

    const int tiles = B_ * (HW_ / 16);             // 65536 waves, 8 per block
    semiconv_wmma_f32<<<tiles / 8, 256, 0, stream>>>(feat, convw, convb, gate, colour);
    genesis_steps<<<B_, 1024, 0, stream>>>(colour, rp, lsig, (float*)d_out, steps);
}